// ModulePSBS_36094905156097
// MI455X (gfx1250) — compile-verified
//
#include <hip/hip_runtime.h>

#define B_DIM 16384
#define F_DIM 256
#define L_DIM 256
#define A_DIM 128

typedef __attribute__((ext_vector_type(2))) float v2f;
typedef __attribute__((ext_vector_type(8))) float v8f;

// ---------------------------------------------------------------------------
// Kernel 1: compose the whole 256-layer mesh into a single complex 256x256
// matrix W, where row k = network(e_k). Each workgroup owns 8 basis rows in
// LDS and sweeps the 256 layers with barriers between butterfly/atten phases.
// W is written TRANSPOSED as W_t[f][k] so the GEMM's B fragments (two
// consecutive K values for a fixed output column f) are single b64 loads.
// ---------------------------------------------------------------------------
#define ROWS_PER_WG 8

__global__ __launch_bounds__(256) void compose_kernel(
    const float* __restrict__ params,   // (L, A) theta
    const int*   __restrict__ index,    // (L, A, 2)
    const float* __restrict__ split,    // (L, A)
    const float* __restrict__ atten_r,  // (L, F)
    const float* __restrict__ atten_p,  // (L, F)
    float* __restrict__ Wre_t,          // (F, F) [f][k]  (transposed)
    float* __restrict__ Wim_t)
{
    __shared__ float yre[ROWS_PER_WG][F_DIM];
    __shared__ float yim[ROWS_PER_WG][F_DIM];

    const int tid  = threadIdx.x;
    const int row0 = blockIdx.x * ROWS_PER_WG;

    // init: rows are basis vectors e_{row0+r}
#pragma unroll
    for (int m = 0; m < (ROWS_PER_WG * F_DIM) / 256; ++m) {
        int w = tid + m * 256;
        int r = w >> 8;
        int f = w & 255;
        yre[r][f] = (row0 + r == f) ? 1.0f : 0.0f;
        yim[r][f] = 0.0f;
    }
    __syncthreads();

    const float PI4 = 0.78539816339744830962f;

    for (int l = 0; l < L_DIM; ++l) {
        // ---- butterfly phase: ROWS*A = 1024 items -> 4 per thread ----
#pragma unroll
        for (int m = 0; m < (ROWS_PER_WG * A_DIM) / 256; ++m) {
            int w = tid + m * 256;
            int r = w >> 7;      // row within workgroup
            int a = w & 127;     // pair index
            int i  = index[(l * A_DIM + a) * 2 + 0];
            int j  = index[(l * A_DIM + a) * 2 + 1];
            float th = params[l * A_DIM + a];
            float s  = PI4 + split[l * A_DIM + a];
            float t  = cosf(s);
            float rr = sinf(s);
            float ct = cosf(th);
            float st = sinf(th);

            float xre = yre[r][i], xim = yim[r][i];
            float ure = xre * ct - xim * st;       // u = x_i * e^{i th}
            float uim = xre * st + xim * ct;
            float vre = yre[r][j], vim = yim[r][j];

            float nire = t * ure - rr * vim;       // ni = t*u + i*r*v
            float niim = t * uim + rr * vre;
            float njre = -rr * uim + t * vre;      // nj = i*r*u + t*v
            float njim =  rr * ure + t * vim;

            yre[r][i] = nire; yim[r][i] = niim;
            yre[r][j] = njre; yim[r][j] = njim;
        }
        __syncthreads();

        // ---- attenuation phase: ROWS*F = 2048 items -> 8 per thread ----
#pragma unroll
        for (int m = 0; m < (ROWS_PER_WG * F_DIM) / 256; ++m) {
            int w = tid + m * 256;
            int r = w >> 8;
            int f = w & 255;
            float ar  = atten_r[l * F_DIM + f];
            float ap  = atten_p[l * F_DIM + f];
            float are = ar * cosf(ap);
            float aim = ar * sinf(ap);
            float xre = yre[r][f], xim = yim[r][f];
            yre[r][f] = xre * are - xim * aim;
            yim[r][f] = xre * aim + xim * are;
        }
        __syncthreads();
    }

    // write out W transposed: W_t[f][row0+r]. Map w so 8 consecutive threads
    // share one f and cover r=0..7 -> 32B-contiguous stores.
#pragma unroll
    for (int m = 0; m < (ROWS_PER_WG * F_DIM) / 256; ++m) {
        int w = tid + m * 256;
        int r = w & 7;
        int f = w >> 3;
        Wre_t[(size_t)f * F_DIM + row0 + r] = yre[r][f];
        Wim_t[(size_t)f * F_DIM + row0 + r] = yim[r][f];
    }
}

// ---------------------------------------------------------------------------
// Kernel 2: complex GEMM  out = X * W  using V_WMMA_F32_16X16X4_F32.
// Each wave computes a 16(M batch) x 64(N feature) slab: 4 N-tiles, each with
// re/im f32 accumulators. F32 WMMA has no A/B negation, so the imaginary B
// fragment is negated in VALU (v_xor sign bit) for the real chain.
//   out_re = Xre*Wre - Xim*Wim ; out_im = Xre*Wim + Xim*Wre
// W is stored transposed [f][k], so every fragment load (A and B) is b64.
// ---------------------------------------------------------------------------
#define N_TILES 4

__global__ __launch_bounds__(256) void cgemm_wmma_kernel(
    const float* __restrict__ Xre,     // (B, F)
    const float* __restrict__ Xim,     // (B, F)
    const float* __restrict__ Wre_t,   // (F, F) [f][k]
    const float* __restrict__ Wim_t,   // (F, F) [f][k]
    float* __restrict__ out)           // (B, F, 2) interleaved re/im
{
    const int wave = threadIdx.x >> 5;          // 0..7
    const int lane = threadIdx.x & 31;

    const int m0 = (blockIdx.x * 8 + wave) * 16;   // batch-row tile base
    const int n0 = blockIdx.y * (N_TILES * 16);    // feature tile base

    // A fragment layout: lanes 0-15 -> M=lane, K={k,k+1}; lanes 16-31 -> K={k+2,k+3}
    const int mrow = lane & 15;
    const int koff = (lane >> 4) * 2;
    // B fragment layout: lanes 0-15 -> K=k   N=lane&15; lanes 16-31 -> K=k+2
    const int ncol = lane & 15;

    v8f cre[N_TILES], cim[N_TILES];
#pragma unroll
    for (int t = 0; t < N_TILES; ++t) { cre[t] = (v8f){}; cim[t] = (v8f){}; }

    const float* arow = Xre + (size_t)(m0 + mrow) * F_DIM + koff;
    const float* brow = Xim + (size_t)(m0 + mrow) * F_DIM + koff;

    // per-tile B base pointers (hoisted; inner loop is base + k)
    const float* bre_p[N_TILES];
    const float* bim_p[N_TILES];
#pragma unroll
    for (int t = 0; t < N_TILES; ++t) {
        const size_t nrow = (size_t)(n0 + t * 16 + ncol) * F_DIM + koff;
        bre_p[t] = Wre_t + nrow;
        bim_p[t] = Wim_t + nrow;
    }

#pragma unroll 2
    for (int k = 0; k < F_DIM; k += 4) {
        v2f a_re = *(const v2f*)(arow + k);
        v2f a_im = *(const v2f*)(brow + k);

#pragma unroll
        for (int t = 0; t < N_TILES; ++t) {
            v2f b_re = *(const v2f*)(bre_p[t] + k);
            v2f b_im = *(const v2f*)(bim_p[t] + k);
            v2f b_imn;
            b_imn.x = -b_im.x;
            b_imn.y = -b_im.y;

            cre[t] = __builtin_amdgcn_wmma_f32_16x16x4_f32(
                false, a_re, false, b_re,  (short)0, cre[t], false, false);
            cre[t] = __builtin_amdgcn_wmma_f32_16x16x4_f32(
                false, a_im, false, b_imn, (short)0, cre[t], false, false);
            cim[t] = __builtin_amdgcn_wmma_f32_16x16x4_f32(
                false, a_re, false, b_im,  (short)0, cim[t], false, false);
            cim[t] = __builtin_amdgcn_wmma_f32_16x16x4_f32(
                false, a_im, false, b_re,  (short)0, cim[t], false, false);
        }
    }

    // C/D layout: VGPR g, lanes 0-15 -> M=g, N=lane; lanes 16-31 -> M=g+8
    const int Mbase = m0 + ((lane >> 4) << 3);
    const int Ncol  = lane & 15;
#pragma unroll
    for (int g = 0; g < 8; ++g) {
#pragma unroll
        for (int t = 0; t < N_TILES; ++t) {
            size_t o = ((size_t)(Mbase + g) * F_DIM + (n0 + t * 16 + Ncol)) * 2;
            v2f val;
            val.x = cre[t][g];
            val.y = cim[t][g];
            *(v2f*)(out + o) = val;   // interleaved {re, im} -> single b64 store
        }
    }
}

// ---------------------------------------------------------------------------
extern "C" void kernel_launch(void* const* d_in, const int* in_sizes, int n_in,
                              void* d_out, int out_size, void* d_ws, size_t ws_size,
                              hipStream_t stream) {
    const float* x_re    = (const float*)d_in[0];
    const float* x_im    = (const float*)d_in[1];
    const float* params  = (const float*)d_in[2];
    const int*   index   = (const int*)d_in[3];
    const float* split   = (const float*)d_in[4];
    const float* atten_r = (const float*)d_in[5];
    const float* atten_p = (const float*)d_in[6];

    float* Wre_t = (float*)d_ws;                   // 256*256 floats
    float* Wim_t = Wre_t + (size_t)F_DIM * F_DIM;  // 256*256 floats (512 KB total)

    compose_kernel<<<F_DIM / ROWS_PER_WG, 256, 0, stream>>>(
        params, index, split, atten_r, atten_p, Wre_t, Wim_t);

    dim3 grid(B_DIM / (8 * 16), F_DIM / (N_TILES * 16));  // (128, 4)
    cgemm_wmma_kernel<<<grid, 256, 0, stream>>>(x_re, x_im, Wre_t, Wim_t, (float*)d_out);
}